// MyParticleNetwork_24678882082840
// MI455X (gfx1250) — compile-verified
//
#include <hip/hip_runtime.h>
#include <math.h>

typedef __attribute__((ext_vector_type(2))) float v2f;
typedef __attribute__((ext_vector_type(8))) float v8f;

#define KBINS 64
#define NREAL 20000

// ---------------------------------------------------------------- utilities
__global__ void k_zero(float* p, long long count) {
  long long i = (long long)blockIdx.x * blockDim.x + threadIdx.x;
  long long stride = (long long)gridDim.x * blockDim.x;
  for (; i < count; i += stride) p[i] = 0.0f;
}

__global__ void k_build_f(const float* __restrict__ feats, float* __restrict__ f, int n) {
  int i = blockIdx.x * blockDim.x + threadIdx.x;
  if (i >= n * 13) return;
  int node = i / 13, c = i % 13;
  f[i] = (c == 0) ? 1.0f : feats[node * 12 + (c - 1)];
}

__global__ void k_relu(const float* __restrict__ x, float* __restrict__ t, long long count) {
  long long i = (long long)blockIdx.x * blockDim.x + threadIdx.x;
  if (i < count) t[i] = fmaxf(x[i], 0.0f);
}

__global__ void k_addinto(float* __restrict__ y, const float* __restrict__ x, long long count) {
  long long i = (long long)blockIdx.x * blockDim.x + threadIdx.x;
  if (i < count) y[i] += x[i];
}

__global__ void k_final(const float* __restrict__ y, float* __restrict__ out, int count) {
  int i = blockIdx.x * blockDim.x + threadIdx.x;
  if (i < count) out[i] = y[i] * (1.0f / 128.0f);
}

// zero-pad a [Kd x cin_cols] weight into [Kd x 64]
__global__ void k_padw(const float* __restrict__ w, float* __restrict__ wpad,
                       int Kd, int cols) {
  int i = blockIdx.x * blockDim.x + threadIdx.x;
  if (i >= Kd * 64) return;
  int r = i >> 6, c = i & 63;
  wpad[i] = (c < cols) ? w[r * cols + c] : 0.0f;
}

// ---------------------------------------------------------------- geometry
__device__ __forceinline__ float signf(float v) {
  return (v > 0.0f) ? 1.0f : ((v < 0.0f) ? -1.0f : 0.0f);
}

__global__ void k_geom(const float* __restrict__ pos, const int* __restrict__ esrc,
                       const int* __restrict__ edst, uint2* __restrict__ kfp,
                       float* __restrict__ wgt, int E, float invR) {
  int e = blockIdx.x * blockDim.x + threadIdx.x;
  if (e >= E) return;
  int s = esrc[e], d = edst[e];
  float x = (pos[d * 3 + 0] - pos[s * 3 + 0]) * invR;
  float y = (pos[d * 3 + 1] - pos[s * 3 + 1]) * invR;
  float z = (pos[d * 3 + 2] - pos[s * 3 + 2]) * invR;
  float r2 = x * x + y * y + z * z;
  float om = 1.0f - r2;
  float win = fminf(fmaxf(om * om * om, 0.0f), 1.0f);

  const float eps = 1e-9f;
  float sq = r2;
  float nrm = sqrtf(sq + eps);
  float rxy2 = x * x + y * y;
  bool polar = 1.25f * z * z > rxy2;
  float sc = polar ? sqrtf(3.0f * nrm / (nrm + fabsf(z) + eps)) : nrm / sqrtf(rxy2 + eps);
  float xc = x * sc, yc = y * sc;
  float zc = polar ? signf(z) * nrm : 1.5f * z;
  if (sq < 1e-12f) { xc = 0.0f; yc = 0.0f; zc = 0.0f; }
  float rr = sqrtf(xc * xc + yc * yc + eps);
  bool c1 = fabsf(xc) >= fabsf(yc);
  float xs = (fabsf(xc) < eps) ? eps : xc;
  float ys = (fabsf(yc) < eps) ? eps : yc;
  const float fop = 1.27323954473516268615f;  // 4/pi
  float a = c1 ? signf(xc) * rr : signf(yc) * rr * fop * atanf(xc / ys);
  float b = c1 ? signf(xc) * rr * fop * atanf(yc / xs) : signf(yc) * rr;
  if (xc * xc + yc * yc < 1e-12f) { a = 0.0f; b = 0.0f; }

  float g[3];
  g[0] = fminf(fmaxf((a * 0.5f + 0.5f) * 3.0f, 0.0f), 3.0f);
  g[1] = fminf(fmaxf((b * 0.5f + 0.5f) * 3.0f, 0.0f), 3.0f);
  g[2] = fminf(fmaxf((zc * 0.5f + 0.5f) * 3.0f, 0.0f), 3.0f);

  int f0[3]; float t3[3];
  #pragma unroll
  for (int ax = 0; ax < 3; ax++) {
    float fl = floorf(g[ax]);
    t3[ax] = g[ax] - fl;
    f0[ax] = (int)fl;
  }

  unsigned lo = 0u, hi = 0u;
  float w8[8];
  #pragma unroll
  for (int c = 0; c < 8; c++) {
    float w = win;
    int kf = 0;
    #pragma unroll
    for (int ax = 0; ax < 3; ax++) {
      int bit = (c >> (2 - ax)) & 1;
      int ia = min(f0[ax] + bit, 3);
      w *= bit ? t3[ax] : (1.0f - t3[ax]);
      kf = kf * 4 + ia;
    }
    w8[c] = w;
    if (c < 4) lo |= ((unsigned)kf) << (8 * c);
    else       hi |= ((unsigned)kf) << (8 * (c - 4));
  }
  kfp[e] = make_uint2(lo, hi);
  float4* wo = (float4*)(wgt + (long long)e * 8);
  wo[0] = make_float4(w8[0], w8[1], w8[2], w8[3]);
  wo[1] = make_float4(w8[4], w8[5], w8[6], w8[7]);
}

// ---------------------------------------------------------------- scatter
__global__ __launch_bounds__(256)
void k_scatter(const float* __restrict__ inp, int cin, const int* __restrict__ esrc,
               const int* __restrict__ edst, const uint2* __restrict__ kfp,
               const float* __restrict__ wgt, float* __restrict__ B, int E) {
  int lane = threadIdx.x & 31;
  int wid = threadIdx.x >> 5;
  int e = blockIdx.x * (blockDim.x >> 5) + wid;
  if (e >= E) return;
  int s = esrc[e], d = edst[e];
  uint2 k8 = kfp[e];
  const float* fr = inp + (long long)d * cin;
  float f0 = (lane < cin) ? fr[lane] : 0.0f;
  float f1 = (cin > 32 && lane + 32 < cin) ? fr[lane + 32] : 0.0f;
  float f2 = (cin > 64 && lane + 64 < cin) ? fr[lane + 64] : 0.0f;
  const float* w8 = wgt + (long long)e * 8;
  long long rowbase = (long long)s * KBINS;
  #pragma unroll
  for (int c = 0; c < 8; c++) {
    unsigned kfc = (((c < 4) ? k8.x : k8.y) >> (8 * (c & 3))) & 63u;
    float w = w8[c];
    if (w != 0.0f) {
      float* bp = B + (rowbase + kfc) * cin;
      if (lane < cin)
        __hip_atomic_fetch_add(bp + lane, f0 * w, __ATOMIC_RELAXED, __HIP_MEMORY_SCOPE_AGENT);
      if (cin > 32 && lane + 32 < cin)
        __hip_atomic_fetch_add(bp + lane + 32, f1 * w, __ATOMIC_RELAXED, __HIP_MEMORY_SCOPE_AGENT);
      if (cin > 64 && lane + 64 < cin)
        __hip_atomic_fetch_add(bp + lane + 64, f2 * w, __ATOMIC_RELAXED, __HIP_MEMORY_SCOPE_AGENT);
    }
  }
}

// ---------------------------------------------------------------- WMMA GEMM
// out[m, o] = bias[o] + sum_r A[m, r] * W64[r, o]   (W pre-padded to 64 cols)
// 256 threads = 8 wave32s (2x4); tile M=32, N=64, K-step 32.
// Double-buffered LDS filled with global_load_async_to_lds_b128 (ASYNCcnt).
#define BM 32
#define BN 64
#define BKT 32
#define APAD 4   // row stride 36 dwords: 16B aligned, conflict-free
#define WPAD 4   // row stride 68 dwords: 16B aligned, conflict-free

__global__ __launch_bounds__(256)
void k_gemm_wmma(const float* __restrict__ A, const float* __restrict__ W,
                 const float* __restrict__ bias, float* __restrict__ out,
                 int n, int Kdim, int cout, int ldo) {
  __shared__ float As[2][BM][BKT + APAD];
  __shared__ float Ws[2][BKT][BN + WPAD];
  int tid = threadIdx.x;
  int lane = tid & 31, wid = tid >> 5;
  int wm = wid >> 2, wn = wid & 3;  // 2 x 4 wave grid
  int l16 = lane & 15, khalf = lane >> 4;
  int m0 = blockIdx.x * BM;

  // per-thread async-copy assignments (uniform, unguarded: A has row slack,
  // W is padded to 64 columns)
  int ar = tid >> 3, ac4 = (tid & 7) * 4;   // A tile: 32 rows x 8 float4
  int wr = tid >> 4, wc4 = (tid & 15) * 4;  // W tile: rows wr, wr+16
  const float* aP = A + (long long)(m0 + ar) * Kdim + ac4;
  const float* wP0 = W + (long long)wr * BN + wc4;
  const float* wP1 = W + (long long)(wr + 16) * BN + wc4;

  auto issue = [&](int kk, int buf) {
    unsigned la  = (unsigned)(uintptr_t)&As[buf][ar][ac4];
    unsigned lw0 = (unsigned)(uintptr_t)&Ws[buf][wr][wc4];
    unsigned lw1 = (unsigned)(uintptr_t)&Ws[buf][wr + 16][wc4];
    unsigned long long ga  = (unsigned long long)(uintptr_t)(aP + kk);
    unsigned long long gw0 = (unsigned long long)(uintptr_t)(wP0 + (long long)kk * BN);
    unsigned long long gw1 = (unsigned long long)(uintptr_t)(wP1 + (long long)kk * BN);
    asm volatile("global_load_async_to_lds_b128 %0, %1, off" :: "v"(la),  "v"(ga)  : "memory");
    asm volatile("global_load_async_to_lds_b128 %0, %1, off" :: "v"(lw0), "v"(gw0) : "memory");
    asm volatile("global_load_async_to_lds_b128 %0, %1, off" :: "v"(lw1), "v"(gw1) : "memory");
  };

  v8f acc = {};
  int mrow = wm * 16 + l16;
  int ncol = wn * 16 + l16;

  issue(0, 0);
  int buf = 0;
  for (int kk = 0; kk < Kdim; kk += BKT) {
    bool hasNext = (kk + BKT) < Kdim;
    if (hasNext) {
      issue(kk + BKT, buf ^ 1);
      // current tile's 3 ops done when <=3 remain (async loads retire in order)
      asm volatile("s_wait_asynccnt 0x3" ::: "memory");
    } else {
      asm volatile("s_wait_asynccnt 0x0" ::: "memory");
    }
    __syncthreads();

    #pragma unroll
    for (int k4 = 0; k4 < BKT; k4 += 4) {
      int ka = k4 + 2 * khalf;  // lanes 0-15: K=k,k+1 ; lanes 16-31: K=k+2,k+3
      v2f av, bv;
      av.x = As[buf][mrow][ka];  av.y = As[buf][mrow][ka + 1];
      bv.x = Ws[buf][ka][ncol];  bv.y = Ws[buf][ka + 1][ncol];
      acc = __builtin_amdgcn_wmma_f32_16x16x4_f32(false, av, false, bv,
                                                  (short)0, acc, false, false);
    }
    __syncthreads();  // buffer free for reuse two tiles ahead
    buf ^= 1;
  }

  int colg = wn * 16 + l16;
  if (colg < cout) {
    float bb = bias[colg];
    #pragma unroll
    for (int v = 0; v < 8; v++) {
      int rowg = m0 + wm * 16 + v + 8 * khalf;  // D: M = vgpr + 8*(lane>=16)
      if (rowg < n) out[(long long)rowg * ldo + colg] = acc[v] + bb;
    }
  }
}

// ---------------------------------------------------------------- dense matvec
__global__ void k_dense(const float* __restrict__ inp, const float* __restrict__ W,
                        const float* __restrict__ b, float* __restrict__ out,
                        int n, int cin, int cout, int ldo, int accumulate) {
  int i = blockIdx.x * blockDim.x + threadIdx.x;
  if (i >= n * cout) return;
  int node = i / cout, o = i % cout;
  const float* ip = inp + (long long)node * cin;
  float v = b[o];
  for (int c = 0; c < cin; c++) v += ip[c] * W[c * cout + o];
  float* op = out + (long long)node * ldo + o;
  if (accumulate) *op += v; else *op = v;
}

// ---------------------------------------------------------------- launch
extern "C" void kernel_launch(void* const* d_in, const int* in_sizes, int n_in,
                              void* d_out, int out_size, void* d_ws, size_t ws_size,
                              hipStream_t stream) {
  const float* pos = (const float*)d_in[0];
  const float* feats = (const float*)d_in[1];
  const int* esrc = (const int*)d_in[2];
  const int* edst = (const int*)d_in[3];
  const float* conv0_w = (const float*)d_in[4];
  const float* conv0_b = (const float*)d_in[5];
  const float* dense0_w = (const float*)d_in[6];
  const float* dense0_b = (const float*)d_in[7];
  const float* conv1_w = (const float*)d_in[8];
  const float* conv1_b = (const float*)d_in[9];
  const float* dense1_w = (const float*)d_in[10];
  const float* dense1_b = (const float*)d_in[11];
  const float* conv2_w = (const float*)d_in[12];
  const float* conv2_b = (const float*)d_in[13];
  const float* dense2_w = (const float*)d_in[14];
  const float* dense2_b = (const float*)d_in[15];
  const float* conv3_w = (const float*)d_in[16];
  const float* conv3_b = (const float*)d_in[17];
  const float* dense3_w = (const float*)d_in[18];
  const float* dense3_b = (const float*)d_in[19];

  const int n = in_sizes[0] / 3;   // 20001
  const int E = in_sizes[2];       // 1200000
  const int npad = (n + BM - 1) & ~(BM - 1);  // row slack for unguarded async A loads
  const float invR = 1.0f / 0.1125f;          // RADIUS = 1.5*6*0.025/2

  char* ws = (char*)d_ws;
  size_t off = 0;
  auto alloc = [&](size_t bytes) -> char* {
    char* p = ws + off;
    off = (off + bytes + 255) & ~(size_t)255;
    return p;
  };
  uint2* kfp = (uint2*)alloc((size_t)E * 8);
  float* wgt = (float*)alloc((size_t)E * 8 * 4);
  float* fbuf = (float*)alloc((size_t)n * 13 * 4);
  float* xbuf = (float*)alloc((size_t)n * 96 * 4);
  float* ybuf = (float*)alloc((size_t)n * 96 * 4);
  float* tbuf = (float*)alloc((size_t)n * 96 * 4);
  float* w3pad = (float*)alloc((size_t)KBINS * 64 * 64 * 4);  // conv3_w padded to 64 cols
  float* Bbuf = (float*)alloc((size_t)npad * KBINS * 96 * 4);
  (void)ws_size;

  const int TB = 256;
  dim3 blk(TB);
  int edgeBlocks = (E + 7) / 8;
  int gemmBlocks = (n + BM - 1) / BM;

  k_geom<<<(E + TB - 1) / TB, blk, 0, stream>>>(pos, esrc, edst, kfp, wgt, E, invR);
  k_build_f<<<(n * 13 + TB - 1) / TB, blk, 0, stream>>>(feats, fbuf, n);
  k_padw<<<(KBINS * 64 * 64 + TB - 1) / TB, blk, 0, stream>>>(conv3_w, w3pad, KBINS * 64, 3);

  // ---- Layer 0: cin=13 -> conv 64 | dense 32 -> x (n x 96)
  {
    k_zero<<<4096, blk, 0, stream>>>(Bbuf, (long long)n * KBINS * 13);
    k_scatter<<<edgeBlocks, blk, 0, stream>>>(fbuf, 13, esrc, edst, kfp, wgt, Bbuf, E);
    k_gemm_wmma<<<gemmBlocks, blk, 0, stream>>>(Bbuf, conv0_w, conv0_b, xbuf,
                                                n, KBINS * 13, 64, 96);
    k_dense<<<(n * 32 + TB - 1) / TB, blk, 0, stream>>>(fbuf, dense0_w, dense0_b,
                                                        xbuf + 64, n, 13, 32, 96, 0);
  }
  float* xp = xbuf;
  float* yp = ybuf;

  // ---- Layer 1: relu(x96) -> conv 96->64 + dense 96->64
  {
    k_relu<<<(int)(((long long)n * 96 + TB - 1) / TB), blk, 0, stream>>>(xp, tbuf, (long long)n * 96);
    k_zero<<<4096, blk, 0, stream>>>(Bbuf, (long long)n * KBINS * 96);
    k_scatter<<<edgeBlocks, blk, 0, stream>>>(tbuf, 96, esrc, edst, kfp, wgt, Bbuf, E);
    k_gemm_wmma<<<gemmBlocks, blk, 0, stream>>>(Bbuf, conv1_w, conv1_b, yp,
                                                n, KBINS * 96, 64, 64);
    k_dense<<<(n * 64 + TB - 1) / TB, blk, 0, stream>>>(tbuf, dense1_w, dense1_b,
                                                        yp, n, 96, 64, 64, 1);
    float* tmp = xp; xp = yp; yp = tmp;
  }

  // ---- Layer 2: residual, 64->64
  {
    k_relu<<<(int)(((long long)n * 64 + TB - 1) / TB), blk, 0, stream>>>(xp, tbuf, (long long)n * 64);
    k_zero<<<4096, blk, 0, stream>>>(Bbuf, (long long)n * KBINS * 64);
    k_scatter<<<edgeBlocks, blk, 0, stream>>>(tbuf, 64, esrc, edst, kfp, wgt, Bbuf, E);
    k_gemm_wmma<<<gemmBlocks, blk, 0, stream>>>(Bbuf, conv2_w, conv2_b, yp,
                                                n, KBINS * 64, 64, 64);
    k_dense<<<(n * 64 + TB - 1) / TB, blk, 0, stream>>>(tbuf, dense2_w, dense2_b,
                                                        yp, n, 64, 64, 64, 1);
    k_addinto<<<(int)(((long long)n * 64 + TB - 1) / TB), blk, 0, stream>>>(yp, xp, (long long)n * 64);
    float* tmp = xp; xp = yp; yp = tmp;
  }

  // ---- Layer 3: 64->3 (weights pre-padded to 64 cols; stores guarded)
  {
    k_relu<<<(int)(((long long)n * 64 + TB - 1) / TB), blk, 0, stream>>>(xp, tbuf, (long long)n * 64);
    k_zero<<<4096, blk, 0, stream>>>(Bbuf, (long long)n * KBINS * 64);
    k_scatter<<<edgeBlocks, blk, 0, stream>>>(tbuf, 64, esrc, edst, kfp, wgt, Bbuf, E);
    k_gemm_wmma<<<gemmBlocks, blk, 0, stream>>>(Bbuf, w3pad, conv3_b, yp,
                                                n, KBINS * 64, 3, 3);
    k_dense<<<(n * 3 + TB - 1) / TB, blk, 0, stream>>>(tbuf, dense3_w, dense3_b,
                                                       yp, n, 64, 3, 3, 1);
  }

  k_final<<<(NREAL * 3 + TB - 1) / TB, blk, 0, stream>>>(yp, (float*)d_out, NREAL * 3);
}